// MAMIL_45586782880051
// MI455X (gfx1250) — compile-verified
//
#include <hip/hip_runtime.h>
#include <hip/hip_bf16.h>

typedef __attribute__((ext_vector_type(2))) float v2f;
typedef __attribute__((ext_vector_type(8))) float v8f;

#define WMMA_F32X4(A_, B_, C_) \
  __builtin_amdgcn_wmma_f32_16x16x4_f32(false, (A_), false, (B_), (short)0, (C_), false, false)

// ---------------- workspace layout (float offsets) ----------------
// pool1 region is dead after conv2, so later tensors alias it.
static constexpr size_t OFF_POOL1 = 0;               // 2048*36*14*14 = 14,450,688
static constexpr size_t OFF_FEAT  = 14450688;        // 2048*1728     =  3,538,944
static constexpr size_t OFF_H1    = 0;               // 2048*512 (aliases pool1)
static constexpr size_t OFF_H     = 1048576;         // 2048*512
static constexpr size_t OFF_AN    = 2097152;         // 2048*512
static constexpr size_t OFF_H2    = 3145728;         // 2048*1024
static constexpr size_t OFF_P     = 5242880;         // 2048*1024
static constexpr size_t OFF_SC    = 7340032;         // 2048*10
static constexpr size_t OFF_BET   = 7360512;         // 10*2048
static constexpr size_t OFF_EMB   = 7380992;         // 10*1024

// ================= conv1 (4x4 VALID) + relu + 2x2 maxpool =================
// x: [2048,32,32] -> pool1: [2048,36,14,14]
__global__ __launch_bounds__(256) void conv1_pool_kernel(
    const float* __restrict__ x, const float* __restrict__ w1,
    const float* __restrict__ b1, float* __restrict__ pool1) {
  const int b = blockIdx.x, tid = threadIdx.x;
  __shared__ float s_img[1024];
  __shared__ float s_w[36 * 16];
  __shared__ float s_b[36];
  for (int i = tid; i < 1024; i += 256) s_img[i] = x[(size_t)b * 1024 + i];
  for (int i = tid; i < 576; i += 256) s_w[i] = w1[i];
  if (tid < 36) s_b[tid] = b1[tid];
  __syncthreads();
  for (int o = tid; o < 36 * 14 * 14; o += 256) {
    const int c = o / 196, rr = o % 196, py = rr / 14, px = rr % 14;
    const float* w = &s_w[c * 16];
    float mx = -1e30f;
    #pragma unroll
    for (int sy = 0; sy < 2; ++sy)
      #pragma unroll
      for (int sx = 0; sx < 2; ++sx) {
        const int y0 = 2 * py + sy, x0 = 2 * px + sx;
        float a = s_b[c];
        #pragma unroll
        for (int i = 0; i < 4; ++i)
          #pragma unroll
          for (int j = 0; j < 4; ++j)
            a += s_img[(y0 + i) * 32 + x0 + j] * w[i * 4 + j];
        mx = fmaxf(mx, fmaxf(a, 0.f));
      }
    pool1[(size_t)b * 7056 + o] = mx;
  }
}

// ========== conv2 (3x3 VALID) as implicit GEMM with WMMA, +relu+pool ==========
// pool1: [2048,36,14,14] -> feat: [2048,1728]  (48*6*6 channel-major flatten)
// GEMM per image: A = w2 [48 x 324], B = im2col [324 x 144], D = [48 x 144]
// im2col is never materialized: a 324-entry LDS offset table maps K -> input
// offset, so the inner loop is ds_load + add (no div/mod on the WMMA path).
__global__ __launch_bounds__(256) void conv2_wmma_kernel(
    const float* __restrict__ pool1, const float* __restrict__ w2,
    const float* __restrict__ b2, float* __restrict__ feat) {
  const int b = blockIdx.x, tid = threadIdx.x;
  __shared__ float s_in[36 * 14 * 14];   // 7056 floats
  __shared__ float s_out[48 * 144];      // 6912 floats (pre-pool, relu'd)
  __shared__ int   s_kofs[324];          // K -> c*196 + dy*14 + dx
  for (int i = tid; i < 7056; i += 256) s_in[i] = pool1[(size_t)b * 7056 + i];
  for (int i = tid; i < 324; i += 256) {
    const int c = i / 9, r = i % 9;
    s_kofs[i] = c * 196 + (r / 3) * 14 + (r % 3);
  }
  __syncthreads();

  const int wave = tid >> 5, lane = tid & 31;
  const int half = lane >> 4, l16 = lane & 15;
  // 3 M-tiles (48 ch) x 9 N-tiles (144 pos) = 27 tiles, round-robin over 8 waves
  for (int t = wave; t < 27; t += 8) {
    const int tm = t / 9, tn = t % 9;
    const int m = tm * 16 + l16;           // output channel for A frag
    const int n = tn * 16 + l16;           // output position for B frag
    const int base = (n / 12) * 14 + (n % 12);
    const float* Arow = w2 + (size_t)m * 324;
    v8f acc = {};
    for (int k = 0; k < 324; k += 4) {
      const int kk = k + 2 * half;
      v2f a, bb;
      a.x = Arow[kk];
      a.y = Arow[kk + 1];
      bb.x = s_in[s_kofs[kk] + base];
      bb.y = s_in[s_kofs[kk + 1] + base];
      acc = WMMA_F32X4(a, bb, acc);
    }
    #pragma unroll
    for (int r = 0; r < 8; ++r) {
      const int ch = tm * 16 + r + 8 * half;
      s_out[ch * 144 + n] = fmaxf(acc[r] + b2[ch], 0.f);
    }
  }
  __syncthreads();
  // 2x2 maxpool 12x12 -> 6x6, channel-major flatten to [1728]
  for (int i = tid; i < 1728; i += 256) {
    const int c = i / 36, rr = i % 36, py = rr / 6, px = rr % 6;
    const float* p = &s_out[c * 144 + (2 * py) * 12 + 2 * px];
    feat[(size_t)b * 1728 + i] =
        fmaxf(fmaxf(p[0], p[1]), fmaxf(p[12], p[13]));
  }
}

// ====== WMMA fp32 GEMM: D[M,N] = act(A[M,K] @ W[N,K]^T + bias) ======
// One wave owns a 32x32 output tile: 2 A-frags x 2 B-frags -> 4 WMMAs per
// K-step with 2x register reuse of every operand fragment. ACT: 0=relu, 1=tanh.
// Requires M%32==0, N%32==0, K%4==0.
template <int ACT>
__global__ __launch_bounds__(256) void gemm_wmma_kernel(
    const float* __restrict__ A, const float* __restrict__ W,
    const float* __restrict__ bias, float* __restrict__ D,
    int M, int N, int K) {
  const int gwave = (blockIdx.x * 256 + threadIdx.x) >> 5;
  const int lane = threadIdx.x & 31;
  const int ntn = N >> 5;
  const int tiles = (M >> 5) * ntn;
  if (gwave >= tiles) return;
  const int tm = gwave / ntn, tn = gwave % ntn;
  const int half = lane >> 4, l16 = lane & 15;
  const int m0 = tm * 32 + l16, n0 = tn * 32 + l16;
  const float* A0 = A + (size_t)m0 * K;
  const float* A1 = A0 + (size_t)16 * K;
  const float* W0 = W + (size_t)n0 * K;   // column n of B = row n of W
  const float* W1 = W0 + (size_t)16 * K;
  v8f acc00 = {}, acc01 = {}, acc10 = {}, acc11 = {};
  #pragma unroll 2
  for (int k = 0; k < K; k += 4) {
    const int kk = k + 2 * half;
    v2f a0, a1, b0, b1;
    a0.x = A0[kk]; a0.y = A0[kk + 1];
    a1.x = A1[kk]; a1.y = A1[kk + 1];
    b0.x = W0[kk]; b0.y = W0[kk + 1];
    b1.x = W1[kk]; b1.y = W1[kk + 1];
    acc00 = WMMA_F32X4(a0, b0, acc00);
    acc01 = WMMA_F32X4(a0, b1, acc01);
    acc10 = WMMA_F32X4(a1, b0, acc10);
    acc11 = WMMA_F32X4(a1, b1, acc11);
  }
  const float bn0 = bias[tn * 32 + l16];
  const float bn1 = bias[tn * 32 + 16 + l16];
  #pragma unroll
  for (int r = 0; r < 8; ++r) {
    const int mmA = tm * 32 + r + 8 * half;        // rows of acc0x
    const int mmB = mmA + 16;                      // rows of acc1x
    float v00 = acc00[r] + bn0, v01 = acc01[r] + bn1;
    float v10 = acc10[r] + bn0, v11 = acc11[r] + bn1;
    if (ACT == 0) {
      v00 = fmaxf(v00, 0.f); v01 = fmaxf(v01, 0.f);
      v10 = fmaxf(v10, 0.f); v11 = fmaxf(v11, 0.f);
    } else {
      v00 = tanhf(v00); v01 = tanhf(v01);
      v10 = tanhf(v10); v11 = tanhf(v11);
    }
    D[(size_t)mmA * N + tn * 32 + l16]      = v00;
    D[(size_t)mmA * N + tn * 32 + 16 + l16] = v01;
    D[(size_t)mmB * N + tn * 32 + l16]      = v10;
    D[(size_t)mmB * N + tn * 32 + 16 + l16] = v11;
  }
}

// ============ sparse 8-neighbor attention (Chebyshev<=1 on 32x64 grid) ============
// H:[2048,512], An:[2048,512] -> H2:[2048,1024] = [H | alphas@H]
__global__ __launch_bounds__(256) void nbr_attn_kernel(
    const float* __restrict__ H, const float* __restrict__ An,
    float* __restrict__ H2) {
  const int i = blockIdx.x, tid = threadIdx.x;
  __shared__ float s_hi[512];
  __shared__ float red[256];
  __shared__ int s_nbr[8];
  __shared__ int s_cnt;
  __shared__ float s_alpha[8];
  if (tid == 0) {
    const int r = i >> 6, c = i & 63;   // positions = (idx//64, idx%64)
    int cnt = 0;
    for (int dr = -1; dr <= 1; ++dr)
      for (int dc = -1; dc <= 1; ++dc) {
        if (dr == 0 && dc == 0) continue;
        const int rr = r + dr, cc = c + dc;
        if (rr >= 0 && rr < 32 && cc >= 0 && cc < 64) s_nbr[cnt++] = rr * 64 + cc;
      }
    s_cnt = cnt;  // always >= 3, reference self-fallback never triggers
  }
  for (int d = tid; d < 512; d += 256) s_hi[d] = H[(size_t)i * 512 + d];
  __syncthreads();
  const int cnt = s_cnt;
  float logit[8];
  for (int jn = 0; jn < cnt; ++jn) {
    const float* an = An + (size_t)s_nbr[jn] * 512;
    float p = 0.f;
    for (int d = tid; d < 512; d += 256) p += s_hi[d] * an[d];
    red[tid] = p; __syncthreads();
    for (int s = 128; s > 0; s >>= 1) {
      if (tid < s) red[tid] += red[tid + s];
      __syncthreads();
    }
    logit[jn] = red[0];
    __syncthreads();
  }
  if (tid == 0) {
    float mx = -1e30f;
    for (int jn = 0; jn < cnt; ++jn) mx = fmaxf(mx, logit[jn]);
    float sum = 0.f;
    for (int jn = 0; jn < cnt; ++jn) { const float e = expf(logit[jn] - mx); s_alpha[jn] = e; sum += e; }
    const float inv = 1.f / sum;
    for (int jn = 0; jn < cnt; ++jn) s_alpha[jn] *= inv;
  }
  __syncthreads();
  for (int d = tid; d < 512; d += 256) {
    float acc = 0.f;
    for (int jn = 0; jn < cnt; ++jn) acc += s_alpha[jn] * H[(size_t)s_nbr[jn] * 512 + d];
    H2[(size_t)i * 1024 + 512 + d] = acc;
    H2[(size_t)i * 1024 + d] = s_hi[d];
  }
}

// scores[n,t] = P[n,:] . templates[t,:]   (10 waves per block, one n per block)
__global__ __launch_bounds__(320) void scores_kernel(
    const float* __restrict__ P, const float* __restrict__ T,
    float* __restrict__ S) {
  const int n = blockIdx.x;
  const int t = threadIdx.x >> 5, lane = threadIdx.x & 31;
  const float* p = P + (size_t)n * 1024;
  const float* tm = T + (size_t)t * 1024;
  float acc = 0.f;
  for (int d = lane; d < 1024; d += 32) acc += p[d] * tm[d];
  for (int off = 16; off > 0; off >>= 1) acc += __shfl_down(acc, off, 32);
  if (lane == 0) S[n * 10 + t] = acc;
}

// betas[t,n] = softmax over n of scores[n,t]
__global__ __launch_bounds__(256) void softmax_n_kernel(
    const float* __restrict__ S, float* __restrict__ B) {
  const int t = blockIdx.x, tid = threadIdx.x;
  __shared__ float red[256];
  float mx = -1e30f;
  for (int n = tid; n < 2048; n += 256) mx = fmaxf(mx, S[n * 10 + t]);
  red[tid] = mx; __syncthreads();
  for (int s = 128; s > 0; s >>= 1) { if (tid < s) red[tid] = fmaxf(red[tid], red[tid + s]); __syncthreads(); }
  mx = red[0]; __syncthreads();
  float sum = 0.f;
  for (int n = tid; n < 2048; n += 256) sum += expf(S[n * 10 + t] - mx);
  red[tid] = sum; __syncthreads();
  for (int s = 128; s > 0; s >>= 1) { if (tid < s) red[tid] += red[tid + s]; __syncthreads(); }
  const float inv = 1.f / red[0];
  for (int n = tid; n < 2048; n += 256) B[t * 2048 + n] = expf(S[n * 10 + t] - mx) * inv;
}

// embs[t,d] = sum_n betas[t,n] * H2[n,d]
__global__ __launch_bounds__(256) void embs_kernel(
    const float* __restrict__ B, const float* __restrict__ H2,
    float* __restrict__ E) {
  const int d = blockIdx.x * 256 + threadIdx.x;
  const int t = blockIdx.y;
  float acc = 0.f;
  for (int n = 0; n < 2048; ++n) acc += B[t * 2048 + n] * H2[(size_t)n * 1024 + d];
  E[t * 1024 + d] = acc;
}

// global attention head: out = [Y_prob(2) | Y_hat(1) | A(2048)]
__global__ __launch_bounds__(256) void final_kernel(
    const float* __restrict__ embs, const float* __restrict__ betas,
    const float* __restrict__ gw1, const float* __restrict__ gb1,
    const float* __restrict__ gw2, const float* __restrict__ gb2,
    const float* __restrict__ cw, const float* __restrict__ cb,
    float* __restrict__ out) {
  const int tid = threadIdx.x;
  __shared__ float red[256];
  __shared__ float s_g[10];
  __shared__ float s_gamma[10];
  __shared__ float s_M[1024];
  for (int t = 0; t < 10; ++t) {
    float h = 0.f;
    if (tid < 128) {
      float a = gb1[tid];
      const float* w = gw1 + (size_t)tid * 1024;
      const float* e = embs + (size_t)t * 1024;
      for (int d = 0; d < 1024; ++d) a += e[d] * w[d];
      h = tanhf(a) * gw2[tid];
    }
    red[tid] = h; __syncthreads();
    for (int s = 128; s > 0; s >>= 1) { if (tid < s) red[tid] += red[tid + s]; __syncthreads(); }
    if (tid == 0) s_g[t] = red[0] + gb2[0];
    __syncthreads();
  }
  if (tid == 0) {
    float mx = -1e30f;
    for (int t = 0; t < 10; ++t) mx = fmaxf(mx, s_g[t]);
    float sum = 0.f;
    for (int t = 0; t < 10; ++t) { const float e = expf(s_g[t] - mx); s_gamma[t] = e; sum += e; }
    const float inv = 1.f / sum;
    for (int t = 0; t < 10; ++t) s_gamma[t] *= inv;
  }
  __syncthreads();
  for (int d = tid; d < 1024; d += 256) {
    float m = 0.f;
    for (int t = 0; t < 10; ++t) m += s_gamma[t] * embs[t * 1024 + d];
    s_M[d] = m;
  }
  for (int n = tid; n < 2048; n += 256) {
    float a = 0.f;
    for (int t = 0; t < 10; ++t) a += s_gamma[t] * betas[t * 2048 + n];
    out[3 + n] = a;
  }
  __syncthreads();
  float l[2];
  for (int c = 0; c < 2; ++c) {
    float p = 0.f;
    for (int d = tid; d < 1024; d += 256) p += s_M[d] * cw[c * 1024 + d];
    red[tid] = p; __syncthreads();
    for (int s = 128; s > 0; s >>= 1) { if (tid < s) red[tid] += red[tid + s]; __syncthreads(); }
    l[c] = red[0];
    __syncthreads();
  }
  if (tid == 0) {
    const float p0 = 1.f / (1.f + expf(-(l[0] + cb[0])));
    const float p1 = 1.f / (1.f + expf(-(l[1] + cb[1])));
    out[0] = p0; out[1] = p1;
    out[2] = (p1 > p0) ? 1.f : 0.f;   // argmax, ties -> 0
  }
}

extern "C" void kernel_launch(void* const* d_in, const int* in_sizes, int n_in,
                              void* d_out, int out_size, void* d_ws, size_t ws_size,
                              hipStream_t stream) {
  const float* x    = (const float*)d_in[0];
  const float* w1   = (const float*)d_in[2];
  const float* b1   = (const float*)d_in[3];
  const float* w2   = (const float*)d_in[4];
  const float* b2   = (const float*)d_in[5];
  const float* fc1w = (const float*)d_in[6];
  const float* fc1b = (const float*)d_in[7];
  const float* fc2w = (const float*)d_in[8];
  const float* fc2b = (const float*)d_in[9];
  const float* nbrw = (const float*)d_in[10];
  const float* nbrb = (const float*)d_in[11];
  const float* tmpl = (const float*)d_in[12];
  const float* pw   = (const float*)d_in[13];
  const float* pb   = (const float*)d_in[14];
  const float* gw1  = (const float*)d_in[15];
  const float* gb1  = (const float*)d_in[16];
  const float* gw2  = (const float*)d_in[17];
  const float* gb2  = (const float*)d_in[18];
  const float* cw   = (const float*)d_in[19];
  const float* cb   = (const float*)d_in[20];
  float* ws  = (float*)d_ws;
  float* out = (float*)d_out;

  conv1_pool_kernel<<<2048, 256, 0, stream>>>(x, w1, b1, ws + OFF_POOL1);
  conv2_wmma_kernel<<<2048, 256, 0, stream>>>(ws + OFF_POOL1, w2, b2, ws + OFF_FEAT);

  // fc1: [2048,1728] x [512,1728]^T -> relu -> H1   (64x16 = 1024 tiles)
  gemm_wmma_kernel<0><<<1024 / 8, 256, 0, stream>>>(
      ws + OFF_FEAT, fc1w, fc1b, ws + OFF_H1, 2048, 512, 1728);
  // fc2: -> relu -> H
  gemm_wmma_kernel<0><<<1024 / 8, 256, 0, stream>>>(
      ws + OFF_H1, fc2w, fc2b, ws + OFF_H, 2048, 512, 512);
  // An = tanh(H @ nbr_w^T + b)
  gemm_wmma_kernel<1><<<1024 / 8, 256, 0, stream>>>(
      ws + OFF_H, nbrw, nbrb, ws + OFF_AN, 2048, 512, 512);

  nbr_attn_kernel<<<2048, 256, 0, stream>>>(ws + OFF_H, ws + OFF_AN, ws + OFF_H2);

  // P = tanh(H2 @ proto_w^T + b)   (64x32 = 2048 tiles)
  gemm_wmma_kernel<1><<<2048 / 8, 256, 0, stream>>>(
      ws + OFF_H2, pw, pb, ws + OFF_P, 2048, 1024, 1024);

  scores_kernel<<<2048, 320, 0, stream>>>(ws + OFF_P, tmpl, ws + OFF_SC);
  softmax_n_kernel<<<10, 256, 0, stream>>>(ws + OFF_SC, ws + OFF_BET);
  embs_kernel<<<dim3(4, 10), 256, 0, stream>>>(ws + OFF_BET, ws + OFF_H2, ws + OFF_EMB);
  final_kernel<<<1, 256, 0, stream>>>(ws + OFF_EMB, ws + OFF_BET,
                                      gw1, gb1, gw2, gb2, cw, cb, out);
}